// VAELoss_14645838479503
// MI455X (gfx1250) — compile-verified
//
#include <hip/hip_runtime.h>
#include <hip/hip_bf16.h>

typedef __attribute__((ext_vector_type(2))) float v2f;
typedef __attribute__((ext_vector_type(8))) float v8f;

#define NPTS   2048
#define NTILES 128      // 2048 / 16
#define BS     16

static constexpr float INV_CNT = 1.0f / (16.0f * 2048.0f);

// ---------------------------------------------------------------------------
// Kernel 1: zero the scalar output, set per-(batch,col) running mins to +inf.
// ---------------------------------------------------------------------------
__global__ void chamfer_init_kernel(float* __restrict__ out,
                                    int* __restrict__ minq, int n) {
    int i = blockIdx.x * blockDim.x + threadIdx.x;
    if (i == 0) out[0] = 0.0f;
    if (i < n) minq[i] = 0x7F800000;   // +inf bits
}

// ---------------------------------------------------------------------------
// Kernel 2: tiled pairwise distances via V_WMMA_F32_16X16X4_F32.
//   grid = (16 batches, 8 i-chunks), block = 256 (8 waves).
//   Each wave owns 2 i-tiles (32 pred rows), fused into ONE sweep over all
//   128 j-tiles: per j-tile -> 1 ds_load_b128, 2 WMMA, 1 ds_min_i32.
//   d2 = |p|^2 + |q|^2 - 2 * (p . q), dot computed by WMMA with K=3 padded.
// ---------------------------------------------------------------------------
__global__ __launch_bounds__(256)
void chamfer_wmma_kernel(const float* __restrict__ pred,
                         const float* __restrict__ tgt,
                         float* __restrict__ out,
                         int* __restrict__ minqG) {
    __shared__ float4 sQ[NPTS];     // target points (x,y,z,|q|^2)  32 KB
    __shared__ int    sMinQ[NPTS];  // per-column running min (f32 bits) 8 KB

    const int b     = blockIdx.x;          // batch
    const int chunk = blockIdx.y;          // i-chunk (2 i-tiles per wave)
    const int tid   = threadIdx.x;
    const int lane  = tid & 31;
    const int wave  = tid >> 5;
    const int half  = lane >> 4;           // 0: K=0,1 half | 1: K=2,3 half
    const int l15   = lane & 15;

    const float* Pb = pred + (size_t)b * NPTS * 3;
    const float* Qb = tgt  + (size_t)b * NPTS * 3;

    // Cooperative load of the target cloud + norms into LDS.
    for (int p = tid; p < NPTS; p += 256) {
        float x = Qb[p * 3 + 0];
        float y = Qb[p * 3 + 1];
        float z = Qb[p * 3 + 2];
        sQ[p]    = make_float4(x, y, z, x * x + y * y + z * z);
        sMinQ[p] = 0x7F800000;
    }
    __syncthreads();

    // ---- A fragments for both i-tiles (rows iBase0.. / iBase1..) ----------
    const int iBase0 = (chunk * 16 + wave * 2) * 16;
    const int iBase1 = iBase0 + 16;

    v2f a0, a1;
    float npv0[8], npv1[8];
    {
        const int r0 = iBase0 + l15;
        float x = Pb[r0 * 3 + 0], y = Pb[r0 * 3 + 1], z = Pb[r0 * 3 + 2];
        a0.x = half ? z    : x;
        a0.y = half ? 0.0f : y;
        float t = half ? (z * z) : (x * x + y * y);
        t += __shfl_xor(t, 16, 32);              // |p|^2 of row iBase0+l15
        #pragma unroll
        for (int v = 0; v < 8; ++v)              // row of D element v: v+8*half
            npv0[v] = __shfl(t, v + 8 * half, 32);
    }
    {
        const int r1 = iBase1 + l15;
        float x = Pb[r1 * 3 + 0], y = Pb[r1 * 3 + 1], z = Pb[r1 * 3 + 2];
        a1.x = half ? z    : x;
        a1.y = half ? 0.0f : y;
        float t = half ? (z * z) : (x * x + y * y);
        t += __shfl_xor(t, 16, 32);
        #pragma unroll
        for (int v = 0; v < 8; ++v)
            npv1[v] = __shfl(t, v + 8 * half, 32);
    }

    float rmin0[8], rmin1[8];
    #pragma unroll
    for (int v = 0; v < 8; ++v) { rmin0[v] = __builtin_inff(); rmin1[v] = __builtin_inff(); }

    // ---- single fused sweep over all target tiles -------------------------
    for (int jt = 0; jt < NTILES; ++jt) {
        const int col = jt * 16 + l15;
        const float4 q = sQ[col];

        v2f bf;
        bf.x = half ? q.z  : q.x;
        bf.y = half ? 0.0f : q.y;
        const float nqv = q.w;          // |q|^2 for N = l15 (shared by all v)

        v8f c0 = {}, c1 = {};
        v8f d0 = __builtin_amdgcn_wmma_f32_16x16x4_f32(
            false, a0, false, bf, (short)0, c0, false, false);
        v8f d1 = __builtin_amdgcn_wmma_f32_16x16x4_f32(
            false, a1, false, bf, (short)0, c1, false, false);

        float cmin = __builtin_inff();
        #pragma unroll
        for (int v = 0; v < 8; ++v) {
            float e0 = fmaxf(npv0[v] + nqv - 2.0f * d0[v], 0.0f);
            float e1 = fmaxf(npv1[v] + nqv - 2.0f * d1[v], 0.0f);
            rmin0[v] = fminf(rmin0[v], e0);
            rmin1[v] = fminf(rmin1[v], e1);
            cmin = fminf(cmin, fminf(e0, e1));
        }
        // Column min: lane L covers M-half `half` of both tiles; lane L+16
        // covers the other half and hits the same address. All 32 lanes issue
        // the non-returning LDS int-min (bit order == float order, all >= 0).
        atomicMin(&sMinQ[col], __float_as_int(cmin));
    }

    // ---- row-min reduction across the 16 lanes of each half ---------------
    float sumP = 0.0f;
    #pragma unroll
    for (int v = 0; v < 8; ++v) {
        #pragma unroll
        for (int m = 1; m <= 8; m <<= 1) {
            rmin0[v] = fminf(rmin0[v], __shfl_xor(rmin0[v], m, 32));
            rmin1[v] = fminf(rmin1[v], __shfl_xor(rmin1[v], m, 32));
        }
    }
    if (l15 == 0) {
        #pragma unroll
        for (int v = 0; v < 8; ++v) sumP += rmin0[v] + rmin1[v];
    }

    // Lanes 0 and 16 hold partial sums; fold and emit once per wave.
    sumP += __shfl_xor(sumP, 16, 32);
    if (lane == 0)
        atomicAdd(out, sumP * INV_CNT);

    // Flush this block's column mins to the global per-batch array.
    __syncthreads();
    for (int p = tid; p < NPTS; p += 256)
        atomicMin(&minqG[b * NPTS + p], sMinQ[p]);
}

// ---------------------------------------------------------------------------
// Kernel 3: sum the target-side mins and fold into the scalar output.
// ---------------------------------------------------------------------------
__global__ __launch_bounds__(256)
void chamfer_finalize_kernel(const int* __restrict__ minq,
                             float* __restrict__ out, int n) {
    __shared__ float red[256];
    float s = 0.0f;
    for (int i = blockIdx.x * blockDim.x + threadIdx.x; i < n;
         i += gridDim.x * blockDim.x)
        s += __int_as_float(minq[i]);
    red[threadIdx.x] = s;
    __syncthreads();
    for (int off = 128; off > 0; off >>= 1) {
        if (threadIdx.x < off) red[threadIdx.x] += red[threadIdx.x + off];
        __syncthreads();
    }
    if (threadIdx.x == 0)
        atomicAdd(out, red[0] * INV_CNT);
}

// ---------------------------------------------------------------------------
extern "C" void kernel_launch(void* const* d_in, const int* in_sizes, int n_in,
                              void* d_out, int out_size, void* d_ws, size_t ws_size,
                              hipStream_t stream) {
    const float* pred = (const float*)d_in[0];   // (16, 2048, 3) f32
    const float* tgt  = (const float*)d_in[1];   // (16, 2048, 3) f32
    float* out = (float*)d_out;                  // scalar f32
    int*   minq = (int*)d_ws;                    // 16*2048 ints = 128 KB

    const int nq = BS * NPTS;

    chamfer_init_kernel<<<(nq + 255) / 256, 256, 0, stream>>>(out, minq, nq);

    dim3 grid(BS, 8);
    chamfer_wmma_kernel<<<grid, 256, 0, stream>>>(pred, tgt, out, minq);

    chamfer_finalize_kernel<<<32, 256, 0, stream>>>(minq, out, nq);
}